// GraphNetwork_16071767621699
// MI455X (gfx1250) — compile-verified
//
#include <hip/hip_runtime.h>
#include <hip/hip_bf16.h>

#define N_NODES 100000
#define N_EDGES 1200000
#define D 64
#define BN_EPS 1e-5f

typedef __attribute__((ext_vector_type(2))) float v2f;
typedef __attribute__((ext_vector_type(8))) float v8f;

static __device__ __forceinline__ void atomic_add_f32(float* p, float v) {
    __hip_atomic_fetch_add(p, v, __ATOMIC_RELAXED, __HIP_MEMORY_SCOPE_AGENT);
}

// ---------------- degree / norm precompute ----------------

__global__ void k_deg_init(float* deg) {
    int i = blockIdx.x * blockDim.x + threadIdx.x;
    if (i < N_NODES) deg[i] = 1.0f;  // self-loop
}

__global__ void k_deg_count(const int* __restrict__ dst, float* deg) {
    int e = blockIdx.x * blockDim.x + threadIdx.x;
    if (e < N_EDGES) atomic_add_f32(&deg[dst[e]], 1.0f);
}

__global__ void k_dinv(float* deg) {
    int i = blockIdx.x * blockDim.x + threadIdx.x;
    if (i < N_NODES) deg[i] = rsqrtf(deg[i]);  // deg >= 1 always (self-loop)
}

__global__ void k_norm(const int* __restrict__ src, const int* __restrict__ dst,
                       const float* __restrict__ dinv, float* __restrict__ norm) {
    int e = blockIdx.x * blockDim.x + threadIdx.x;
    if (e < N_EDGES) norm[e] = dinv[src[e]] * dinv[dst[e]];
}

// ---------------- fused fp32 WMMA GEMM ----------------
// hw[n x 64]  = f(A)[n x 64] @ W[64 x 64]           (f = inline BN+ReLU if BN_IN)
// agg[n x 64] = hw * dinv[row]^2 (+ bias)           (self-loop term, initializes agg)
//
// Block = 128 threads = 4 waves; each wave computes a 16-row x 64-col tile.
// V_WMMA_F32_16X16X4_F32: A 16x4 (v2f/lane), B 4x16 (v2f/lane), C/D 16x16 (v8f/lane).
//
// W staged in LDS fragment-major so each lane's B-fragment (K,K+1 pair for its
// column) is one contiguous 8B-aligned ds_load_b64:
//   Wl[kb*256 + j*64 + lane*2 + t] = W[(kb*4 + (lane>>4)*2 + t)*64 + (j*16 + (lane&15))]

template <bool BN_IN>
__global__ __launch_bounds__(128) void k_gemm_fused(
        const float* __restrict__ A, const float* __restrict__ W,
        const float* __restrict__ stats, const float* __restrict__ gamma,
        const float* __restrict__ beta, const float* __restrict__ dinv,
        const float* __restrict__ bias, float* __restrict__ hw,
        float* __restrict__ agg, int n) {
    __shared__ float Wl[64 * 64];
    __shared__ float scl[64], shf[64];

    for (int f = threadIdx.x; f < 64 * 64; f += 128) {
        int t  = f & 1;
        int ln = (f >> 1) & 31;
        int j  = (f >> 6) & 3;
        int kb = f >> 8;
        int kk = kb * 4 + ((ln >> 4) << 1) + t;
        int nn = j * 16 + (ln & 15);
        Wl[f] = W[kk * 64 + nn];
    }
    if (BN_IN) {
        const float inv_n = 1.0f / (float)N_NODES;
        for (int d = threadIdx.x; d < 64; d += 128) {
            float mean = stats[d] * inv_n;
            float var  = stats[D + d] * inv_n - mean * mean;
            float sc   = gamma[d] * rsqrtf(var + BN_EPS);
            scl[d] = sc;
            shf[d] = beta[d] - mean * sc;
        }
    }
    __syncthreads();

    const int wave = threadIdx.x >> 5;
    const int lane = threadIdx.x & 31;
    const int r0   = blockIdx.x * 64 + wave * 16;
    const int m    = lane & 15;           // row-within-tile (A)
    const int kh   = (lane >> 4) << 1;    // K sub-offset: 0 (lanes 0-15) or 2 (lanes 16-31)

    int arow = r0 + m;
    if (arow > n - 1) arow = n - 1;       // clamp (stores are guarded)
    const float* ap = A + (size_t)arow * D;
    const float* bp = Wl + lane * 2;      // per-lane fragment base

    v8f acc0 = {}, acc1 = {}, acc2 = {}, acc3 = {};

#pragma unroll
    for (int kb = 0; kb < 16; ++kb) {
        const int k = kb * 4;
        v2f a;
        a.x = ap[k + kh];
        a.y = ap[k + kh + 1];
        if (BN_IN) {  // inline BatchNorm(train)+ReLU on the input activation
            a.x = fmaxf(a.x * scl[k + kh]     + shf[k + kh],     0.0f);
            a.y = fmaxf(a.y * scl[k + kh + 1] + shf[k + kh + 1], 0.0f);
        }
        v2f b0 = *(const v2f*)(bp + kb * 256);
        v2f b1 = *(const v2f*)(bp + kb * 256 + 64);
        v2f b2 = *(const v2f*)(bp + kb * 256 + 128);
        v2f b3 = *(const v2f*)(bp + kb * 256 + 192);
        acc0 = __builtin_amdgcn_wmma_f32_16x16x4_f32(false, a, false, b0, (short)0, acc0, false, false);
        acc1 = __builtin_amdgcn_wmma_f32_16x16x4_f32(false, a, false, b1, (short)0, acc1, false, false);
        acc2 = __builtin_amdgcn_wmma_f32_16x16x4_f32(false, a, false, b2, (short)0, acc2, false, false);
        acc3 = __builtin_amdgcn_wmma_f32_16x16x4_f32(false, a, false, b3, (short)0, acc3, false, false);
    }

    // C/D layout: VGPR v -> M=v (lanes 0-15), M=v+8 (lanes 16-31); N = lane&15.
    const int rbase = r0 + ((lane >> 4) << 3);
    const int c     = lane & 15;
    float bb0 = 0.0f, bb1 = 0.0f, bb2 = 0.0f, bb3 = 0.0f;
    if (bias) {
        bb0 = bias[c];
        bb1 = bias[c + 16];
        bb2 = bias[c + 32];
        bb3 = bias[c + 48];
    }
#pragma unroll
    for (int v = 0; v < 8; ++v) {
        int row = rbase + v;
        if (row < n) {
            float di = dinv[row];
            float s  = di * di;
            float* hp = hw  + (size_t)row * D + c;
            float* gp = agg + (size_t)row * D + c;
            hp[0]  = acc0[v];
            hp[16] = acc1[v];
            hp[32] = acc2[v];
            hp[48] = acc3[v];
            gp[0]  = acc0[v] * s + bb0;
            gp[16] = acc1[v] * s + bb1;
            gp[32] = acc2[v] * s + bb2;
            gp[48] = acc3[v] * s + bb3;
        }
    }
}

// ---------------- edge aggregation ----------------
// 64 consecutive threads per edge: coalesced gather of hw[src] and coalesced atomics on agg[dst].
__global__ void k_edge_scatter(const float* __restrict__ hw, const int* __restrict__ src,
                               const int* __restrict__ dst, const float* __restrict__ norm,
                               float* __restrict__ agg) {
    long long i = (long long)blockIdx.x * blockDim.x + threadIdx.x;
    if (i < (long long)N_EDGES * D) {
        int e = (int)(i >> 6), d = (int)(i & 63);
        float v = hw[(size_t)src[e] * D + d] * norm[e];
        atomic_add_f32(&agg[(size_t)dst[e] * D + d], v);
    }
}

// ---------------- batchnorm statistics ----------------

__global__ void k_zero_stats(float* stats) {
    if (threadIdx.x < 2 * D) stats[threadIdx.x] = 0.0f;
}

// stats[0:64] = sum per channel, stats[64:128] = sumsq per channel.
__global__ __launch_bounds__(256) void k_bn_stats(const float* __restrict__ h,
                                                  float* __restrict__ stats, int n) {
    __shared__ float ls[256], lss[256];
    int c = threadIdx.x & 63, g = threadIdx.x >> 6;
    int rbeg = blockIdx.x * 256;
    int rend = rbeg + 256; if (rend > n) rend = n;
    float s = 0.0f, ss = 0.0f;
    for (int r = rbeg + g; r < rend; r += 4) {
        float v = h[(size_t)r * D + c];
        s += v; ss += v * v;
    }
    ls[threadIdx.x] = s; lss[threadIdx.x] = ss;
    __syncthreads();
    if (g == 0) {
        float S  = ls[c]  + ls[64 + c]  + ls[128 + c]  + ls[192 + c];
        float SS = lss[c] + lss[64 + c] + lss[128 + c] + lss[192 + c];
        atomic_add_f32(&stats[c], S);
        atomic_add_f32(&stats[D + c], SS);
    }
}

// ---------------- launch ----------------

extern "C" void kernel_launch(void* const* d_in, const int* in_sizes, int n_in,
                              void* d_out, int out_size, void* d_ws, size_t ws_size,
                              hipStream_t stream) {
    const float* x      = (const float*)d_in[0];
    const int*   eidx   = (const int*)d_in[1];     // [2, E]
    const float* W1     = (const float*)d_in[3];
    // b1 (d_in[4]) cancels in train-mode BN — skipped (exact).
    const float* gamma1 = (const float*)d_in[5];
    const float* beta1  = (const float*)d_in[6];
    const float* W2     = (const float*)d_in[7];
    // b2 (d_in[8]) cancels too.
    const float* gamma2 = (const float*)d_in[9];
    const float* beta2  = (const float*)d_in[10];
    const float* W3     = (const float*)d_in[11];
    const float* b3     = (const float*)d_in[12];

    const int* src = eidx;            // edge_index[0]
    const int* dst = eidx + N_EDGES;  // edge_index[1]

    float* w = (float*)d_ws;
    const size_t NF = (size_t)N_NODES * D;  // 6.4M floats
    float* dinv  = w;                       // N (deg -> dinv in place)
    float* norm  = dinv + 100096;           // E
    float* hw    = norm + N_EDGES;          // N*64  (GEMM output, gather source)
    float* aggA  = hw + NF;                 // N*64  (aggregated, layer 1)
    float* aggB  = aggA + NF;               // N*64  (aggregated, layer 2)
    float* stats = aggB + NF;               // 128

    float* outp = (float*)d_out;

    const int gN    = (N_NODES + 255) / 256;                        // 391
    const int gE    = (N_EDGES + 255) / 256;                        // 4688
    const int gED   = (int)(((long long)N_EDGES * D + 255) / 256);  // 300000
    const int gGemm = (N_NODES + 63) / 64;                          // 1563

    // --- normalization precompute ---
    k_deg_init <<<gN, 256, 0, stream>>>(dinv);
    k_deg_count<<<gE, 256, 0, stream>>>(dst, dinv);
    k_dinv     <<<gN, 256, 0, stream>>>(dinv);
    k_norm     <<<gE, 256, 0, stream>>>(src, dst, dinv, norm);

    // --- layer 1: GEMM(+selfloop epilogue) -> scatter -> stats ---
    k_gemm_fused<false><<<gGemm, 128, 0, stream>>>(
        x, W1, nullptr, nullptr, nullptr, dinv, nullptr, hw, aggA, N_NODES);
    k_edge_scatter<<<gED, 256, 0, stream>>>(hw, src, dst, norm, aggA);
    k_zero_stats  <<<1, 128, 0, stream>>>(stats);
    k_bn_stats    <<<gN, 256, 0, stream>>>(aggA, stats, N_NODES);

    // --- layer 2: GEMM(BN+ReLU prologue, selfloop epilogue) -> scatter -> stats ---
    k_gemm_fused<true><<<gGemm, 128, 0, stream>>>(
        aggA, W2, stats, gamma1, beta1, dinv, nullptr, hw, aggB, N_NODES);
    k_edge_scatter<<<gED, 256, 0, stream>>>(hw, src, dst, norm, aggB);
    k_zero_stats  <<<1, 128, 0, stream>>>(stats);   // safe: layer-2 GEMM already consumed stats
    k_bn_stats    <<<gN, 256, 0, stream>>>(aggB, stats, N_NODES);

    // --- layer 3: GEMM(BN+ReLU prologue, selfloop+bias epilogue -> d_out) -> scatter ---
    k_gemm_fused<true><<<gGemm, 128, 0, stream>>>(
        aggB, W3, stats, gamma2, beta2, dinv, b3, hw, outp, N_NODES);
    k_edge_scatter<<<gED, 256, 0, stream>>>(hw, src, dst, norm, outp);
}